// MambaBlock_40183714021464
// MI455X (gfx1250) — compile-verified
//
#include <hip/hip_runtime.h>

typedef float v2f __attribute__((ext_vector_type(2)));
typedef float v8f __attribute__((ext_vector_type(8)));

#define D_MODEL 1024
#define D_INNER 4096
#define N_STATE 16
#define DT_RANK 64
#define SEQ     1024
#define BATCH   2
#define BL      (BATCH * SEQ)              // 2048 rows
#define XDBL_N  (DT_RANK + 2 * N_STATE)    // 96
#define KCHUNK  64
#define KSTEPS  (KCHUNK / 4)               // 16 WMMA K-steps per chunk

__device__ __forceinline__ float sigmoidf_(float x) {
    return 1.0f / (1.0f + __expf(-x));
}

// ---------------------------------------------------------------------------
// f32 WMMA GEMM: C[M,N] = A[M,K] @ B[K,N] (row major, arbitrary ld)
// Requirements: M % 256 == 0, N % 16 == 0, K % 64 == 0.
// Block = 256 threads (8 waves) owns a 256(M) x 16(N) tile:
//   - B K-chunk (64x16 = 4KB) staged in LDS, double-buffered, coalesced
//     float4 loads; WMMA B fragments via ds_load_2addr.
//   - A K-chunk fragments staged into REGISTERS up front (one clause of
//     32 global_load_b64) so WMMAs drain against a decreasing loadcnt
//     instead of stalling to 0 every step.
//   - Each wave computes two 16x16 tiles (shared B fragment) with
//     V_WMMA_F32_16X16X4_F32.
// epilogue: 0 = none, 1 = softplus
// ---------------------------------------------------------------------------
__global__ __launch_bounds__(256) void wmma_gemm_f32(
    const float* __restrict__ A, const float* __restrict__ B,
    float* __restrict__ C,
    int M, int N, int K, int lda, int ldb, int ldc, int epilogue)
{
    __shared__ __align__(16) float Bs[2][KCHUNK][16];

    const int tid   = threadIdx.x;
    const int lane  = tid & 31;
    const int wv    = tid >> 5;                 // wave in block: 0..7
    const int ntile = N >> 4;
    const int bm    = blockIdx.x / ntile;       // 256-row block index
    const int tn    = blockIdx.x - bm * ntile;  // N tile index
    const int half  = lane >> 4;                // which K-pair this half-wave holds
    const int l16   = lane & 15;

    // Per-wave A row streams for the two M tiles (offset by this lane's K-pair).
    const float* __restrict__ Arow0 =
        A + (size_t)(bm * 256 + wv * 32 + l16) * lda + 2 * half;
    const float* __restrict__ Arow1 = Arow0 + (size_t)16 * lda;

    // Cooperative B chunk load: 64 rows x 16 cols, one float4 per thread.
    const int brow = tid >> 2;                  // 0..63
    const int bcol = (tid & 3) << 2;            // 0,4,8,12
    const float* __restrict__ Bld = B + (size_t)brow * ldb + tn * 16 + bcol;
    const size_t bstep = (size_t)KCHUNK * ldb;

    v8f acc0 = {0.f, 0.f, 0.f, 0.f, 0.f, 0.f, 0.f, 0.f};
    v8f acc1 = {0.f, 0.f, 0.f, 0.f, 0.f, 0.f, 0.f, 0.f};

    const int nchunk = K / KCHUNK;

    // Prologue: stage B chunk 0.
    {
        float4 b0 = *(const float4*)Bld;
        *(float4*)&Bs[0][brow][bcol] = b0;
    }
    __syncthreads();

    for (int c = 0; c < nchunk; ++c) {
        const int  cur  = c & 1;
        const int  nxt  = cur ^ 1;
        const bool more = (c + 1) < nchunk;      // block-uniform

        // Issue next B chunk's global loads early so they hide under WMMA.
        float4 bnext;
        if (more) bnext = *(const float4*)(Bld + (size_t)(c + 1) * bstep);

        // Stage this chunk's A fragments into registers: one batch of loads.
        const float* __restrict__ pA0 = Arow0 + c * KCHUNK;
        const float* __restrict__ pA1 = Arow1 + c * KCHUNK;
        v2f a0r[KSTEPS], a1r[KSTEPS];
#pragma unroll
        for (int i = 0; i < KSTEPS; ++i) {
            a0r[i] = *(const v2f*)(pA0 + 4 * i);
            a1r[i] = *(const v2f*)(pA1 + 4 * i);
        }
        __builtin_prefetch(pA0 + KCHUNK, 0, 3);  // global_prefetch_b8
        __builtin_prefetch(pA1 + KCHUNK, 0, 3);

        // Matrix loop: LDS B fragments + register A fragments.
#pragma unroll
        for (int i = 0; i < KSTEPS; ++i) {
            v2f b;
            b.x = Bs[cur][4 * i + 2 * half][l16];
            b.y = Bs[cur][4 * i + 2 * half + 1][l16];
            acc0 = __builtin_amdgcn_wmma_f32_16x16x4_f32(
                       false, a0r[i], false, b, (short)0, acc0, false, false);
            acc1 = __builtin_amdgcn_wmma_f32_16x16x4_f32(
                       false, a1r[i], false, b, (short)0, acc1, false, false);
        }

        if (more) *(float4*)&Bs[nxt][brow][bcol] = bnext;
        __syncthreads();
    }

    // Epilogue: write both 16x16 tiles.
    const int col   = tn * 16 + l16;
    const int rowA0 = bm * 256 + wv * 32 + half * 8;
    const int rowA1 = rowA0 + 16;
#pragma unroll
    for (int i = 0; i < 8; ++i) {
        float v0 = acc0[i];
        float v1 = acc1[i];
        if (epilogue == 1) {                    // softplus
            v0 = (v0 > 20.0f) ? v0 : log1pf(expf(v0));
            v1 = (v1 > 20.0f) ? v1 : log1pf(expf(v1));
        }
        C[(size_t)(rowA0 + i) * ldc + col] = v0;
        C[(size_t)(rowA1 + i) * ldc + col] = v1;
    }
}

// ---------------------------------------------------------------------------
// Depthwise causal conv (4 taps along L) + bias + SiLU.
// xz is (B, L, 2*D_INNER); the conv input xc is columns [0, D_INNER).
// ---------------------------------------------------------------------------
__global__ __launch_bounds__(256) void conv_silu_k(
    const float* __restrict__ xz, const float* __restrict__ Wconv,
    const float* __restrict__ bconv, float* __restrict__ u)
{
    const int gid = blockIdx.x * blockDim.x + threadIdx.x;   // BL * D_INNER
    const int d   = gid & (D_INNER - 1);
    const int bt  = gid >> 12;            // D_INNER == 2^12
    const int t   = bt & (SEQ - 1);
    const int b   = bt >> 10;             // SEQ == 2^10

    float acc = bconv[d];
    const float* __restrict__ w = Wconv + d * 4;
#pragma unroll
    for (int k = 0; k < 4; ++k) {
        const int tt = t - 3 + k;
        if (tt >= 0)
            acc += xz[((size_t)(b * SEQ + tt)) * (2 * D_INNER) + d] * w[k];
    }
    u[gid] = acc * sigmoidf_(acc);        // SiLU
}

// ---------------------------------------------------------------------------
// Selective scan. One lane per (d, n) pair: 16 lanes cooperate per channel.
// h[t] = exp(dt*A)*h[t-1] + dt*B[t]*u[t];  y[t] = sum_n h*C[t]
// Fused epilogue: y = (y + u*D) * silu(res).  y aliases delta (each lane-group
// reads delta[t] before writing y[t] at the same address).
// ---------------------------------------------------------------------------
__global__ __launch_bounds__(256) void mamba_scan_k(
    const float* __restrict__ delta, const float* __restrict__ u,
    const float* __restrict__ xdbl,  const float* __restrict__ xz,
    const float* __restrict__ A_log, const float* __restrict__ Dvec,
    float* __restrict__ y)
{
    const int gid = blockIdx.x * blockDim.x + threadIdx.x;  // BATCH*D_INNER*16
    const int n   = gid & (N_STATE - 1);
    const int dg  = gid >> 4;
    const int d   = dg & (D_INNER - 1);
    const int b   = dg >> 12;

    const float Adn = -expf(A_log[d * N_STATE + n]);
    float h = 0.0f;
    const size_t rowBase = (size_t)b * SEQ;

    for (int t = 0; t < SEQ; ++t) {
        const size_t r  = rowBase + t;
        const float dt = delta[r * D_INNER + d];
        const float uu = u[r * D_INNER + d];
        const float Bn = xdbl[r * XDBL_N + DT_RANK + n];
        const float Cn = xdbl[r * XDBL_N + DT_RANK + N_STATE + n];

        h = expf(dt * Adn) * h + dt * Bn * uu;   // dt>=0, Adn<=0 -> stable
        float yv = h * Cn;

        yv += __shfl_xor(yv, 8, 16);
        yv += __shfl_xor(yv, 4, 16);
        yv += __shfl_xor(yv, 2, 16);
        yv += __shfl_xor(yv, 1, 16);

        if (n == 0) {
            const float res = xz[r * (2 * D_INNER) + D_INNER + d];
            const float out = (yv + uu * Dvec[d]) * (res * sigmoidf_(res));
            y[r * D_INNER + d] = out;
        }
    }
}

// ---------------------------------------------------------------------------
extern "C" void kernel_launch(void* const* d_in, const int* in_sizes, int n_in,
                              void* d_out, int out_size, void* d_ws, size_t ws_size,
                              hipStream_t stream)
{
    const float* x     = (const float*)d_in[0];  // (2,1024,1024)
    const float* Win   = (const float*)d_in[1];  // (1024, 8192)
    const float* Wconv = (const float*)d_in[2];  // (4096, 4)
    const float* bconv = (const float*)d_in[3];  // (4096,)
    const float* Wx    = (const float*)d_in[4];  // (4096, 96)
    const float* Wdt   = (const float*)d_in[5];  // (64, 4096)
    const float* A_log = (const float*)d_in[6];  // (4096, 16)
    const float* Dv    = (const float*)d_in[7];  // (4096,)
    const float* Wout  = (const float*)d_in[8];  // (4096, 1024)
    float* out = (float*)d_out;                  // (2,1024,1024)

    float* ws    = (float*)d_ws;
    float* xz    = ws;                                   // 2048 * 8192
    float* u     = xz   + (size_t)BL * 2 * D_INNER;      // 2048 * 4096
    float* xdbl  = u    + (size_t)BL * D_INNER;          // 2048 * 96
    float* delta = xdbl + (size_t)BL * XDBL_N;           // 2048 * 4096
    float* y     = delta;                                // alias (safe, see scan)

    // 1) xz = x @ Win                       (2048 x 8192, K=1024)
    {
        const int M = BL, N = 2 * D_INNER, K = D_MODEL;
        wmma_gemm_f32<<<(M / 256) * (N / 16), 256, 0, stream>>>(
            x, Win, xz, M, N, K, K, N, N, 0);
    }
    // 2) u = silu(causal_conv(xc) + bconv)
    conv_silu_k<<<(BL * D_INNER) / 256, 256, 0, stream>>>(xz, Wconv, bconv, u);

    // 3) x_dbl = u @ Wx                     (2048 x 96, K=4096)
    {
        const int M = BL, N = XDBL_N, K = D_INNER;
        wmma_gemm_f32<<<(M / 256) * (N / 16), 256, 0, stream>>>(
            u, Wx, xdbl, M, N, K, K, N, N, 0);
    }
    // 4) delta = softplus(x_dbl[:, :64] @ Wdt)   (2048 x 4096, K=64, lda=96)
    {
        const int M = BL, N = D_INNER, K = DT_RANK;
        wmma_gemm_f32<<<(M / 256) * (N / 16), 256, 0, stream>>>(
            xdbl, Wdt, delta, M, N, K, XDBL_N, N, N, 1);
    }
    // 5) selective scan + fused (u*D) and silu(res) gating -> y (aliases delta)
    mamba_scan_k<<<(BATCH * D_INNER * N_STATE) / 256, 256, 0, stream>>>(
        delta, u, xdbl, xz, A_log, Dv, y);

    // 6) out = y @ Wout                     (2048 x 1024, K=4096)
    {
        const int M = BL, N = D_MODEL, K = D_INNER;
        wmma_gemm_f32<<<(M / 256) * (N / 16), 256, 0, stream>>>(
            y, Wout, out, M, N, K, K, N, N, 0);
    }
}